// OVDBaseClassifier_41858751267083
// MI455X (gfx1250) — compile-verified
//
#include <hip/hip_runtime.h>
#include <math.h>

// ---------------------------------------------------------------------------
// Problem constants (from reference)
// ---------------------------------------------------------------------------
#define B_SZ   2
#define K_SZ   1849          // patches = 43*43
#define D_SZ   1024
#define N_SZ   128
#define KP     1856          // K padded to multiple of 16 (116 tiles)
#define MTILES 116           // ceil(1849/16)
#define PATCH  43
#define TGT    602
#define OCH    152           // output-row chunk per workgroup (4 chunks cover 602)
#define EPSN   1e-8f

typedef float v2f __attribute__((ext_vector_type(2)));
typedef float v4f __attribute__((ext_vector_type(4)));
typedef float v8f __attribute__((ext_vector_type(8)));

// ---------------------------------------------------------------------------
// Kernel 0: row L2 norms of feats (B*K rows) and embeddings (N rows).
// One block per row; 256-thread LDS tree reduction.
// ---------------------------------------------------------------------------
__global__ __launch_bounds__(256)
void ovd_norms_kernel(const float* __restrict__ feats,
                      const float* __restrict__ emb,
                      float* __restrict__ fn,     // [B][KP]
                      float* __restrict__ en) {   // [N]
    __shared__ float red[256];
    const int r = blockIdx.x;
    const float* src;
    float* dst;
    if (r < B_SZ * K_SZ) {
        const int b = r / K_SZ;
        const int k = r - b * K_SZ;
        src = feats + (size_t)r * D_SZ;
        dst = fn + (size_t)b * KP + k;
    } else {
        const int j = r - B_SZ * K_SZ;
        src = emb + (size_t)j * D_SZ;
        dst = en + j;
    }
    float s = 0.0f;
    for (int i = threadIdx.x; i < D_SZ; i += 256) {
        float v = src[i];
        s += v * v;
    }
    red[threadIdx.x] = s;
    __syncthreads();
    for (int off = 128; off > 0; off >>= 1) {
        if (threadIdx.x < off) red[threadIdx.x] += red[threadIdx.x + off];
        __syncthreads();
    }
    if (threadIdx.x == 0) *dst = sqrtf(red[0]);
}

// ---------------------------------------------------------------------------
// Kernel 1: S[b][n][k] = (feats[b,k,:] . emb[n,:]) / (|feats_k||emb_n| + eps)
// fp32 WMMA GEMM: V_WMMA_F32_16X16X4_F32, D = A(16x4) x B(4x16) + C.
//   - workgroup = one 16-row M-tile (k-dim), blockIdx.y = batch
//   - 8 waves, wave w owns N-tile [16w, 16w+16)
//   - reduction over D=1024 in K=4 steps -> 256 chained WMMAs
// Per ISA layouts (CDNA5 ISA):
//   A 16x4 f32 : lane<16 -> M=lane, holds K={0,1}; lane>=16 -> same M, K={2,3}
//   B 4x16 f32 : lane<16 -> N=lane, holds K={0,1}; lane>=16 -> same N, K={2,3}
//   => both fragments are one contiguous float2 per lane (b64 loads).
//   C 16x16 f32: vgpr j, lane<16 -> (M=j, N=lane); lane>=16 -> (M=j+8, N=lane-16)
// EXEC must be all-1s at the WMMA: loads are clamped, stores are masked.
// Epilogue: full tiles (115/116) store two aligned b128s per lane.
// ---------------------------------------------------------------------------
__global__ __launch_bounds__(256)
void ovd_gemm_wmma_kernel(const float* __restrict__ feats,
                          const float* __restrict__ emb,
                          const float* __restrict__ fn,
                          const float* __restrict__ en,
                          const int*   __restrict__ normalize,
                          float* __restrict__ S) {
    const int mbase = blockIdx.x * 16;
    const int b     = blockIdx.y;
    const int lane  = threadIdx.x & 31;
    const int wave  = threadIdx.x >> 5;
    const int half  = lane >> 4;        // 0 -> K pair {0,1}, 1 -> {2,3}
    const int l15   = lane & 15;

    int rowA = mbase + l15;
    if (rowA > K_SZ - 1) rowA = K_SZ - 1;           // clamp (masked at store)
    const int n = wave * 16 + l15;

    const v2f* __restrict__ Arow = (const v2f*)(feats + ((size_t)b * K_SZ + rowA) * D_SZ);
    const v2f* __restrict__ Brow = (const v2f*)(emb + (size_t)n * D_SZ);

    v8f c = {0.f, 0.f, 0.f, 0.f, 0.f, 0.f, 0.f, 0.f};
#pragma unroll 8
    for (int kb = 0; kb < D_SZ / 4; ++kb) {
        v2f a  = Arow[kb * 2 + half];   // feats[row, 4kb + 2*half .. +1]
        v2f bb = Brow[kb * 2 + half];   // emb  [n,   4kb + 2*half .. +1]
        // 8 args: (neg_a, A, neg_b, B, c_mod, C, reuse_a, reuse_b)
        c = __builtin_amdgcn_wmma_f32_16x16x4_f32(
                false, a, false, bb, (short)0, c, false, false);
    }

    const int   do_norm = *normalize;
    const float env     = en[n];
    float* Srow = S + (size_t)(b * N_SZ + n) * KP;
    const int mtop = mbase + (half << 3);           // upper halves hold M=j+8

    float vals[8];
#pragma unroll
    for (int j = 0; j < 8; ++j) {
        float v = c[j];
        if (do_norm) v = v / (fn[(size_t)b * KP + (mtop + j)] * env + EPSN);
        vals[j] = v;
    }
    if (mbase + 16 <= K_SZ) {
        // Full tile: mtop is a multiple of 8 -> 32B-aligned b128 stores.
        v4f lo = {vals[0], vals[1], vals[2], vals[3]};
        v4f hi = {vals[4], vals[5], vals[6], vals[7]};
        *(v4f*)(Srow + mtop)     = lo;
        *(v4f*)(Srow + mtop + 4) = hi;
    } else {
#pragma unroll
        for (int j = 0; j < 8; ++j) {
            const int m = mtop + j;
            if (m < K_SZ) Srow[m] = vals[j];
        }
    }
}

// ---------------------------------------------------------------------------
// Kernel 2: separable bicubic 43 -> 602 (exact torch kernel, a = -0.75).
// M has only 4 taps/row, so instead of dense (602x43)x(43x43)x(43x602) GEMMs
// we do a 4-tap gather filter with the intermediate T = M.S held in LDS.
// Output (371 MB) is the roofline term (~16us at 23.3 TB/s); final stores
// are coalesced b64 (602 = 2*301 float2 per row, rows 8B-aligned).
// grid = (B*N maps, 4 o-chunks of 152 rows)
// ---------------------------------------------------------------------------
__device__ __forceinline__ float ovd_cubic_w(float x) {
    const float a = -0.75f;
    float ax = fabsf(x);
    float w1 = ((a + 2.0f) * ax - (a + 3.0f)) * ax * ax + 1.0f;       // |x|<=1
    float w2 = a * (((ax - 5.0f) * ax + 8.0f) * ax - 4.0f);           // 1<|x|<2
    return ax <= 1.0f ? w1 : (ax < 2.0f ? w2 : 0.0f);
}

__global__ __launch_bounds__(256)
void ovd_bicubic_kernel(const float* __restrict__ S,
                        float* __restrict__ out) {
    __shared__ float Slds[PATCH * 44];   // S map, pitch 44 (bank-friendly)
    __shared__ float Tlds[OCH * 44];     // T chunk = M.S rows, pitch 44
    __shared__ float pwgt[TGT * 4];      // column taps (all 602)
    __shared__ int   pidx[TGT * 4];
    __shared__ float owgt[OCH * 4];      // row taps for this chunk
    __shared__ int   oidx[OCH * 4];

    const int bn    = blockIdx.x;                // b*128 + n
    const int o0    = blockIdx.y * OCH;
    const int cnt   = min(OCH, TGT - o0);
    const int tid   = threadIdx.x;

    // Stage the 43x43 similarity map into LDS.
    const float* src = S + (size_t)bn * KP;
    for (int k = tid; k < PATCH * PATCH; k += 256) {
        const int h = k / PATCH;
        const int w = k - h * PATCH;
        Slds[h * 44 + w] = src[k];
    }
    // Bicubic taps for every output position (half-pixel centers, clamped
    // indices; clamped duplicates sum, matching np.add.at in the reference).
    for (int o = tid; o < TGT; o += 256) {
        const float xs = (o + 0.5f) * ((float)PATCH / (float)TGT) - 0.5f;
        const float fl = floorf(xs);
        const float t  = xs - fl;
        const int   i0 = (int)fl;
#pragma unroll
        for (int s = 0; s < 4; ++s) {
            const int off = s - 1;
            float w = ovd_cubic_w(t - (float)off);
            int ix = i0 + off;
            ix = ix < 0 ? 0 : (ix > PATCH - 1 ? PATCH - 1 : ix);
            pwgt[o * 4 + s] = w;
            pidx[o * 4 + s] = ix;
        }
    }
    __syncthreads();
    // Copy this chunk's row taps (small, avoids re-deriving per element).
    for (int ol = tid; ol < cnt; ol += 256) {
        const int o = o0 + ol;
#pragma unroll
        for (int s = 0; s < 4; ++s) {
            owgt[ol * 4 + s] = pwgt[o * 4 + s];
            oidx[ol * 4 + s] = pidx[o * 4 + s];
        }
    }
    __syncthreads();
    // T[ol][w] = sum_t owgt[t] * S[oidx[t]][w]   (vertical pass)
    for (int e = tid; e < cnt * PATCH; e += 256) {
        const int ol = e / PATCH;
        const int w  = e - ol * PATCH;
        const int t4 = ol * 4;
        float acc = owgt[t4 + 0] * Slds[oidx[t4 + 0] * 44 + w]
                  + owgt[t4 + 1] * Slds[oidx[t4 + 1] * 44 + w]
                  + owgt[t4 + 2] * Slds[oidx[t4 + 2] * 44 + w]
                  + owgt[t4 + 3] * Slds[oidx[t4 + 3] * 44 + w];
        Tlds[ol * 44 + w] = acc;
    }
    __syncthreads();
    // out[o][p] = sum_t pwgt[t] * T[ol][pidx[t]]  (horizontal pass)
    // Two consecutive p per thread -> coalesced b64 stores.
    for (int ol = 0; ol < cnt; ++ol) {
        const float* Trow = &Tlds[ol * 44];
        v2f* orow = (v2f*)(out + ((size_t)bn * TGT + (size_t)(o0 + ol)) * TGT);
        for (int pp = tid; pp < TGT / 2; pp += 256) {
            const int t4 = pp * 8;   // taps for p = 2*pp and p = 2*pp+1
            float v0 = pwgt[t4 + 0] * Trow[pidx[t4 + 0]]
                     + pwgt[t4 + 1] * Trow[pidx[t4 + 1]]
                     + pwgt[t4 + 2] * Trow[pidx[t4 + 2]]
                     + pwgt[t4 + 3] * Trow[pidx[t4 + 3]];
            float v1 = pwgt[t4 + 4] * Trow[pidx[t4 + 4]]
                     + pwgt[t4 + 5] * Trow[pidx[t4 + 5]]
                     + pwgt[t4 + 6] * Trow[pidx[t4 + 6]]
                     + pwgt[t4 + 7] * Trow[pidx[t4 + 7]];
            v2f v = {v0, v1};
            orow[pp] = v;
        }
    }
}

// ---------------------------------------------------------------------------
// Launch: norms -> WMMA GEMM (+normalize) -> bicubic upsample.
// Workspace layout (floats): S[B*N*KP] | fn[B*KP] | en[N]  (~1.83 MB)
// ---------------------------------------------------------------------------
extern "C" void kernel_launch(void* const* d_in, const int* in_sizes, int n_in,
                              void* d_out, int out_size, void* d_ws, size_t ws_size,
                              hipStream_t stream) {
    const float* feats     = (const float*)d_in[0];
    const float* emb       = (const float*)d_in[1];
    const int*   normalize = (const int*)d_in[2];

    float* ws = (float*)d_ws;
    float* S  = ws;                                   // B*N*KP
    float* fn = S + (size_t)B_SZ * N_SZ * KP;         // B*KP
    float* en = fn + (size_t)B_SZ * KP;               // N
    float* out = (float*)d_out;

    ovd_norms_kernel<<<dim3(B_SZ * K_SZ + N_SZ), dim3(256), 0, stream>>>(
        feats, emb, fn, en);

    ovd_gemm_wmma_kernel<<<dim3(MTILES, B_SZ), dim3(256), 0, stream>>>(
        feats, emb, fn, en, normalize, S);

    ovd_bicubic_kernel<<<dim3(B_SZ * N_SZ, (TGT + OCH - 1) / OCH), dim3(256), 0, stream>>>(
        S, out);
}